// PCENetwork_19765439496561
// MI455X (gfx1250) — compile-verified
//
#include <hip/hip_runtime.h>
#include <hip/hip_bf16.h>

// ---------------------------------------------------------------------------
// PCE network forward on MI455X (gfx1250, wave32).
// Heavy expert convs run as per-patch implicit GEMMs on V_WMMA_F32_16X16X4_F32
// (full fp32 precision, f32 accumulate).
// ---------------------------------------------------------------------------

typedef float v2f __attribute__((ext_vector_type(2)));
typedef float v8f __attribute__((ext_vector_type(8)));

#define PI_F 3.14159265358979323846f

// ---------------------------------------------------------------------------
// Router: per-patch Fourier-coordinate features -> 128-d embedding -> 8 expert
// logits -> softmax -> threshold(>1/16) -> renormalize.  Purely geometric
// (independent of batch), one wave per patch.
// ---------------------------------------------------------------------------
__global__ __launch_bounds__(32)
void router_kernel(const float* __restrict__ proj_w,   // [128,20]
                   const float* __restrict__ proj_b,   // [128]
                   const float* __restrict__ keys,     // [8,128]
                   float* __restrict__ scores,         // [P,8]
                   int ps, int npx) {
  const int p  = blockIdx.x;
  const int py = p / npx, px = p - py * npx;
  const int t  = threadIdx.x;

  __shared__ float phi[20];
  __shared__ float acc[32][8];

  if (t < 20) {
    float s = 0.0f;
    for (int i = 0; i < ps; ++i) {
      float y = -1.0f + (2.0f / 207.0f) * (float)(py * ps + i);
      for (int j = 0; j < ps; ++j) {
        float x = -1.0f + (2.0f / 207.0f) * (float)(px * ps + j);
        float r = sqrtf(x * x + y * y);
        float a = atan2f(y, x);
        float v;
        if (t == 0)      v = x;
        else if (t == 1) v = y;
        else if (t == 2) v = r;
        else if (t == 3) v = a;
        else {
          int   q    = t - 4;
          float w    = (q < 8) ? 2.0f * PI_F : 4.0f * PI_F;   // f=1 -> 2pi, f=2 -> 4pi
          int   idx  = q & 7;
          int   sel  = idx >> 1;
          float base = (sel == 0) ? x : (sel == 1) ? y : (sel == 2) ? r : a;
          v = (idx & 1) ? cosf(w * base) : sinf(w * base);
        }
        s += v;
      }
    }
    phi[t] = s / (float)(ps * ps);
  }
  __syncthreads();

  float la[8];
#pragma unroll
  for (int k = 0; k < 8; ++k) la[k] = 0.0f;
  for (int d = t; d < 128; d += 32) {
    float e = proj_b[d];
    for (int c = 0; c < 20; ++c) e += phi[c] * proj_w[d * 20 + c];
#pragma unroll
    for (int k = 0; k < 8; ++k) la[k] += e * keys[k * 128 + d];
  }
#pragma unroll
  for (int k = 0; k < 8; ++k) acc[t][k] = la[k];
  __syncthreads();

  if (t == 0) {
    float lg[8];
#pragma unroll
    for (int k = 0; k < 8; ++k) {
      float s = 0.0f;
      for (int l = 0; l < 32; ++l) s += acc[l][k];
      lg[k] = s;
    }
    float mx = lg[0];
#pragma unroll
    for (int k = 1; k < 8; ++k) mx = fmaxf(mx, lg[k]);
    float se = 0.0f;
#pragma unroll
    for (int k = 0; k < 8; ++k) { lg[k] = expf(lg[k] - mx); se += lg[k]; }
    float s2 = 0.0f;
#pragma unroll
    for (int k = 0; k < 8; ++k) {
      lg[k] /= se;
      if (!(lg[k] > 0.0625f)) lg[k] = 0.0f;   // strict > THRESH
      s2 += lg[k];
    }
    float inv = 1.0f / (s2 + 1e-6f);
#pragma unroll
    for (int k = 0; k < 8; ++k) scores[p * 8 + k] = lg[k] * inv;
  }
}

// ---------------------------------------------------------------------------
// Layer 0: ps=16, Cin=3, E*Cout=64.  Per patch: im2col [256 x 28] in LDS,
// weights [28 x 64] in LDS.  4 waves; wave w owns N-tile of 16 channels.
// M: 16 tiles of 16.  K: 7 steps of 4 (f32 WMMA).  Epilogue: bias + relu +
// score, cross-expert combine + fused 1x1 conv (8->8), write to img0.
// ---------------------------------------------------------------------------
__global__ __launch_bounds__(128)
void pce_layer0_kernel(const float* __restrict__ X,       // [16,3,208,208]
                       const float* __restrict__ w_exp,   // [64,27] ([8,8,3,3,3] flat)
                       const float* __restrict__ b_exp,   // [64]
                       const float* __restrict__ w_fc,    // [8,8]
                       const float* __restrict__ b_fc,    // [8]
                       const float* __restrict__ scores,  // [169,8]
                       float* __restrict__ img0)          // [16,8,208,208]
{
  const int blk = blockIdx.x;
  const int b = blk / 169, p = blk - b * 169;
  const int py = p / 13, px = p - py * 13;
  const int t = threadIdx.x;
  const int wv = t >> 5, lane = t & 31;

  __shared__ float As[256][28];          // im2col  (28,672 B)
  __shared__ float Bs[28][64];           // weights ( 7,168 B)
  __shared__ float scr[4][16][16];       // per-wave D tiles
  __shared__ float sco[8], fcb[8];
  __shared__ float fcw[64], bias[64];

  for (int i = t; i < 28 * 64; i += 128) {
    int k = i >> 6, n = i & 63;
    Bs[k][n] = (k < 27) ? w_exp[n * 27 + k] : 0.0f;
  }
  if (t < 64) { fcw[t] = w_fc[t]; bias[t] = b_exp[t]; }
  if (t < 8)  { fcb[t] = b_fc[t]; sco[t] = scores[p * 8 + t]; }

  const float* Xb = X + (size_t)b * 3 * 208 * 208;
  const int oy = py * 16, ox = px * 16;
  for (int i = t; i < 256 * 28; i += 128) {
    int m = i / 28, k = i - m * 28;
    float v = 0.0f;
    if (k < 27) {
      int c = k / 9, r = k - c * 9;
      int ii = (m >> 4) + r / 3 - 1;
      int jj = (m & 15) + r % 3 - 1;
      if ((unsigned)ii < 16u && (unsigned)jj < 16u)
        v = Xb[((size_t)c * 208 + (oy + ii)) * 208 + (ox + jj)];
    }
    As[m][k] = v;
  }
  __syncthreads();

  float* Ob = img0 + (size_t)b * 8 * 208 * 208;
  const int arow = lane & 15;
  const int koff = (lane < 16) ? 0 : 2;            // A/B K-pair per lane half
  const int ncol = (wv << 4) | (lane & 15);        // fused expert-channel oc
  const float sc = sco[ncol >> 3];                 // expert = oc/8
  const float bi = bias[ncol];
  const int rbase = (lane >> 4) << 3;              // D rows: v or v+8

  for (int mt = 0; mt < 16; ++mt) {
    v8f acc = {};
    for (int ks = 0; ks < 7; ++ks) {
      v2f af, bf;
      af.x = As[(mt << 4) | arow][ks * 4 + koff];
      af.y = As[(mt << 4) | arow][ks * 4 + koff + 1];
      bf.x = Bs[ks * 4 + koff][ncol];
      bf.y = Bs[ks * 4 + koff + 1][ncol];
      acc = __builtin_amdgcn_wmma_f32_16x16x4_f32(false, af, false, bf,
                                                  (short)0, acc, false, false);
    }
#pragma unroll
    for (int v = 0; v < 8; ++v)
      scr[wv][rbase + v][lane & 15] = fmaxf(acc[v] + bi, 0.0f) * sc;
    __syncthreads();
    {
      int ml = t >> 3, c = t & 7;                  // 128 threads = 16 rows x 8 ch
      float out = fcb[c];
#pragma unroll
      for (int k = 0; k < 8; ++k) {
        float ck = 0.0f;
#pragma unroll
        for (int w = 0; w < 4; ++w) ck += scr[w][ml][k] + scr[w][ml][k + 8];
        out += fcw[c * 8 + k] * ck;
      }
      int m = (mt << 4) | ml;
      Ob[((size_t)c * 208 + (oy + (m >> 4))) * 208 + (ox + (m & 15))] = out;
    }
    __syncthreads();
  }
}

// ---------------------------------------------------------------------------
// Layer 1: ps=13, Cin=8, E*Cout=128, Cout=16.  Per patch: im2col [176 x 72]
// in LDS (~50 KB), B streamed from global (small, L2/WGP$-resident).
// 8 waves; wave w == expert w (N-tile of 16 = its 16 output channels).
// Fuses: combine over experts, 1x1 conv (16->16), and the 26x26 avg-pool
// partial sum (each 13x13 patch lies wholly inside one pool cell).
// ---------------------------------------------------------------------------
__global__ __launch_bounds__(256)
void pce_layer1_kernel(const float* __restrict__ img0,    // [16,8,208,208]
                       const float* __restrict__ w_exp,   // [128,72] ([8,16,8,3,3] flat)
                       const float* __restrict__ b_exp,   // [128]
                       const float* __restrict__ w_fc,    // [16,16]
                       const float* __restrict__ b_fc,    // [16]
                       const float* __restrict__ scores,  // [256,8]
                       float* __restrict__ partials)      // [16,256,16] patch channel sums
{
  const int blk = blockIdx.x;
  const int b = blk >> 8, p = blk & 255;
  const int py = p >> 4, px = p & 15;
  const int t = threadIdx.x;
  const int wv = t >> 5, lane = t & 31;

  __shared__ float As[176][72];          // 50,688 B
  __shared__ float scr[8][16][16];       //  8,192 B
  __shared__ float etile[16][16];
  __shared__ float sco[8], fcb[16], accc[16];
  __shared__ float fcw[256], bias[128];

  fcw[t] = w_fc[t];                       // 256 threads, 256 entries
  if (t < 128) bias[t] = b_exp[t];
  if (t < 16)  { fcb[t] = b_fc[t]; accc[t] = 0.0f; }
  if (t < 8)   sco[t] = scores[p * 8 + t];

  const float* Ib = img0 + (size_t)b * 8 * 208 * 208;
  const int oy = py * 13, ox = px * 13;
  for (int i = t; i < 176 * 72; i += 256) {
    int m = i / 72, k = i - m * 72;
    float v = 0.0f;
    if (m < 169) {
      int mi = m / 13, mj = m - mi * 13;
      int c = k / 9, r = k - c * 9;
      int ii = mi + r / 3 - 1, jj = mj + r % 3 - 1;
      if ((unsigned)ii < 13u && (unsigned)jj < 13u)
        v = Ib[((size_t)c * 208 + (oy + ii)) * 208 + (ox + jj)];
    }
    As[m][k] = v;
  }
  __syncthreads();

  const int arow = lane & 15;
  const int koff = (lane < 16) ? 0 : 2;
  const int ncol = (wv << 4) | (lane & 15);       // oc = expert*16 + co
  const float sc = sco[wv];                       // expert == wave
  const float bi = bias[ncol];
  const int rbase = (lane >> 4) << 3;
  const float* __restrict__ wrow = w_exp + (size_t)ncol * 72;

  for (int mt = 0; mt < 11; ++mt) {
    v8f acc = {};
    for (int ks = 0; ks < 18; ++ks) {
      v2f af, bf;
      const float* ap = &As[(mt << 4) | arow][ks * 4 + koff];
      af.x = ap[0]; af.y = ap[1];
      const float* bp = wrow + ks * 4 + koff;
      bf.x = bp[0]; bf.y = bp[1];
      acc = __builtin_amdgcn_wmma_f32_16x16x4_f32(false, af, false, bf,
                                                  (short)0, acc, false, false);
    }
#pragma unroll
    for (int v = 0; v < 8; ++v)
      scr[wv][rbase + v][lane & 15] = fmaxf(acc[v] + bi, 0.0f) * sc;
    __syncthreads();
    {
      int ml = t >> 4, c = t & 15;                // 256 threads = 16 rows x 16 ch
      float out = fcb[c];
#pragma unroll
      for (int k = 0; k < 16; ++k) {
        float ck = 0.0f;
#pragma unroll
        for (int w = 0; w < 8; ++w) ck += scr[w][ml][k];
        out += fcw[c * 16 + k] * ck;
      }
      etile[ml][c] = out;
    }
    __syncthreads();
    if (t < 16) {                                  // pool partial: sum valid rows
      float s = 0.0f;
      int mbase = mt << 4;
      for (int ml = 0; ml < 16; ++ml)
        if (mbase + ml < 169) s += etile[ml][t];
      accc[t] += s;
    }
    __syncthreads();
  }
  if (t < 16) partials[((size_t)(b * 256 + p)) * 16 + t] = accc[t];
}

// ---------------------------------------------------------------------------
// Head: assemble pooled [16,16,8,8] (4 patches per pool cell, /676), then
// dense 1024 -> 100 per image.
// ---------------------------------------------------------------------------
__global__ __launch_bounds__(128)
void head_kernel(const float* __restrict__ partials,   // [16,256,16]
                 const float* __restrict__ w_cls,      // [100,1024]
                 const float* __restrict__ b_cls,      // [100]
                 float* __restrict__ out)              // [16,100]
{
  const int b = blockIdx.x;
  const int t = threadIdx.x;
  __shared__ float flat[1024];
  const float* pb = partials + (size_t)b * 256 * 16;
  for (int i = t; i < 1024; i += 128) {
    int c = i >> 6;              // flatten order: c*64 + by*8 + bx
    int r = i & 63;
    int by = r >> 3, bx = r & 7;
    float s = 0.0f;
    for (int dy = 0; dy < 2; ++dy)
      for (int dx = 0; dx < 2; ++dx)
        s += pb[(((2 * by + dy) * 16) + (2 * bx + dx)) * 16 + c];
    flat[i] = s * (1.0f / 676.0f);
  }
  __syncthreads();
  for (int o = t; o < 100; o += 128) {
    float s = b_cls[o];
    for (int i = 0; i < 1024; ++i) s += flat[i] * w_cls[o * 1024 + i];
    out[b * 100 + o] = s;
  }
}

// ---------------------------------------------------------------------------
extern "C" void kernel_launch(void* const* d_in, const int* in_sizes, int n_in,
                              void* d_out, int out_size, void* d_ws, size_t ws_size,
                              hipStream_t stream) {
  const float* X       = (const float*)d_in[0];
  const float* w_exp0  = (const float*)d_in[1];
  const float* b_exp0  = (const float*)d_in[2];
  const float* w_fc0   = (const float*)d_in[3];
  const float* b_fc0   = (const float*)d_in[4];
  const float* proj_w0 = (const float*)d_in[5];
  const float* proj_b0 = (const float*)d_in[6];
  const float* keys0   = (const float*)d_in[7];
  const float* w_exp1  = (const float*)d_in[8];
  const float* b_exp1  = (const float*)d_in[9];
  const float* w_fc1   = (const float*)d_in[10];
  const float* b_fc1   = (const float*)d_in[11];
  const float* proj_w1 = (const float*)d_in[12];
  const float* proj_b1 = (const float*)d_in[13];
  const float* keys1   = (const float*)d_in[14];
  const float* w_cls   = (const float*)d_in[15];
  const float* b_cls   = (const float*)d_in[16];

  // Workspace layout (floats): scores0 | scores1 | img0 | partials  (~22.5 MB)
  float* ws       = (float*)d_ws;
  float* scores0  = ws;                       // 169*8
  float* scores1  = ws + 2048;                // 256*8
  float* img0     = ws + 8192;                // 16*8*208*208 = 5,537,792
  float* partials = img0 + 5537792;           // 16*256*16    =    65,536

  router_kernel<<<169, 32, 0, stream>>>(proj_w0, proj_b0, keys0, scores0, 16, 13);
  router_kernel<<<256, 32, 0, stream>>>(proj_w1, proj_b1, keys1, scores1, 13, 16);
  pce_layer0_kernel<<<16 * 169, 128, 0, stream>>>(X, w_exp0, b_exp0, w_fc0, b_fc0,
                                                  scores0, img0);
  pce_layer1_kernel<<<16 * 256, 256, 0, stream>>>(img0, w_exp1, b_exp1, w_fc1, b_fc1,
                                                  scores1, partials);
  head_kernel<<<16, 128, 0, stream>>>(partials, w_cls, b_cls, (float*)d_out);
}